// FractalAttention_21612275433837
// MI455X (gfx1250) — compile-verified
//
#include <hip/hip_runtime.h>

// FractalAttention on MI455X (gfx1250): per level, fused
//   mh[128,128] = X[128,128] @ W_h[128,128]      (stage A, WMMA f32 16x16x4)
//   y[16,128]   = mh-as-[16,1024] @ w_mix + b    (stage B, WMMA f32 16x16x4)
// One 256-thread (8-wave) workgroup per 16 output rows; mh lives in LDS.

#define BB 8
#define SS 8192
#define DD 128
#define HH 8
#define RTOT (BB * SS)          // 65536 rows
#define MH_STRIDE 132           // LDS row stride (floats), padded vs bank conflicts

typedef float v2f __attribute__((ext_vector_type(2)));
typedef float v8f __attribute__((ext_vector_type(8)));

// ---------------------------------------------------------------------------
// Prep: transpose weights [4,8,128,128] (d,e)->(e,d) and w_mix [1024,128]->[128,1024]
// so every WMMA B-fragment is one contiguous b64 load.
// ---------------------------------------------------------------------------
__global__ void fractal_prep_transpose(const float* __restrict__ weights,
                                       const float* __restrict__ w_mix,
                                       float* __restrict__ weightsT,
                                       float* __restrict__ w_mixT) {
  int idx = blockIdx.x * blockDim.x + threadIdx.x;
  const int total_w = 4 * 8 * 128 * 128;   // 524288
  if (idx < total_w) {
    int m = idx >> 14;          // which [128,128] matrix
    int r = idx & 16383;
    int e = r >> 7;             // output row (e)
    int d = r & 127;            // output col (d)
    weightsT[idx] = weights[(m << 14) + (d << 7) + e];
  }
  const int total_m = 128 * 1024;          // 131072
  if (idx < total_m) {
    int e = idx >> 10;          // output row (e: 0..127)
    int k = idx & 1023;         // output col (k: 0..1023)
    w_mixT[idx] = w_mix[(k << 7) + e];
  }
}

// ---------------------------------------------------------------------------
// One tree level. L = 1<<Lshift (segment length), Lmask = L-1.
// Block gm handles global output rows [16*gm, 16*gm+16).
// ---------------------------------------------------------------------------
__global__ __launch_bounds__(256)
void fractal_level_kernel(const float* __restrict__ xin,
                          const float* __restrict__ weightsT,  // this level: [8][128][128] (e,d)
                          const float* __restrict__ w_mixT,    // [128][1024] (e,k)
                          const float* __restrict__ b_mix,     // [128]
                          float* __restrict__ xout,
                          int Lmask, int Lshift) {
  __shared__ float lds_mh[128 * MH_STRIDE];

  const int gm    = blockIdx.x;        // 0..4095
  const int tid   = threadIdx.x;
  const int wave  = tid >> 5;          // 0..7
  const int lane  = tid & 31;
  const int lhalf = lane >> 4;         // 0/1 (K-pair select in A/B fragments)
  const int lm    = lane & 15;         // M (A) or N (B) within 16

  const int r0       = gm << 4;                 // global output-row base
  const int l0       = (r0 & (SS - 1)) & Lmask; // row offset within segment
  const int seg_base = r0 - l0;                 // global row of segment start
  const int p0       = l0 << 3;                 // mh-row base in [H*L) space
  const int h        = p0 >> Lshift;            // head for this whole tile
  const int lp0      = p0 & Lmask;              // x-row base within segment

  const float* Wh = weightsT + (h << 14);       // [e][d] 128x128, transposed

  // ---------------- Stage A: mh[128][128] = X[128][128] @ W_h ---------------
  {
    v8f acc[8];
#pragma unroll
    for (int j = 0; j < 8; ++j)
#pragma unroll
      for (int v = 0; v < 8; ++v) acc[j][v] = 0.0f;

    const int arow = (wave << 4) + lm;          // this lane's X row in tile
    const float* xr =
        xin + (long)(seg_base + lp0 + arow) * DD + (lhalf << 1);

    for (int k0 = 0; k0 < 128; k0 += 4) {
      v2f a = *(const v2f*)(xr + k0);           // A[lm, k0+2*lhalf+{0,1}]
#pragma unroll
      for (int j = 0; j < 8; ++j) {
        // B[k, n] = W[k][n] = Wh(e=16j+lm, d=k0+2*lhalf+{0,1}) contiguous b64
        v2f bf = *(const v2f*)(Wh + ((j << 4) + lm) * DD + (lhalf << 1) + k0);
        acc[j] = __builtin_amdgcn_wmma_f32_16x16x4_f32(
            false, a, false, bf, (short)0, acc[j], false, false);
      }
    }
    // C[v,lane] = mh[16*wave + 8*lhalf + v][16j + lm] -> LDS row-major
    const int rbase = (wave << 4) + (lhalf << 3);
#pragma unroll
    for (int j = 0; j < 8; ++j) {
      const int col = (j << 4) + lm;
#pragma unroll
      for (int v = 0; v < 8; ++v)
        lds_mh[(rbase + v) * MH_STRIDE + col] = acc[j][v];
    }
  }
  __syncthreads();

  // -------- Stage B: y[16][128] = flat[16][1024] @ w_mix + b_mix ------------
  // flat[i, c*128+e] = mh_local[8i + c][e];  wave owns output cols [16w,16w+16)
  {
    const int n0 = wave << 4;
    const float bias = b_mix[n0 + lm];
    v8f acc;
#pragma unroll
    for (int v = 0; v < 8; ++v) acc[v] = bias;

    const float* wmrow = w_mixT + (n0 + lm) * 1024 + (lhalf << 1);

#pragma unroll
    for (int c = 0; c < 8; ++c) {
      const float* arow =
          &lds_mh[((lm << 3) + c) * MH_STRIDE + (lhalf << 1)];
      const float* brow = wmrow + (c << 7);
      for (int e0 = 0; e0 < 128; e0 += 4) {
        v2f a  = *(const v2f*)(arow + e0);      // ds_load_b64
        v2f bf = *(const v2f*)(brow + e0);      // global_load_b64 (L2-hot)
        acc = __builtin_amdgcn_wmma_f32_16x16x4_f32(
            false, a, false, bf, (short)0, acc, false, false);
      }
    }
    // D[v,lane] = y[8*lhalf + v][n0 + lm]
    float* yout = xout + (long)r0 * DD + n0 + lm;
#pragma unroll
    for (int v = 0; v < 8; ++v)
      yout[(long)((lhalf << 3) + v) * DD] = acc[v];
  }
}

// ---------------------------------------------------------------------------
extern "C" void kernel_launch(void* const* d_in, const int* in_sizes, int n_in,
                              void* d_out, int out_size, void* d_ws, size_t ws_size,
                              hipStream_t stream) {
  const float* x       = (const float*)d_in[0];  // [8,8192,128]
  const float* weights = (const float*)d_in[1];  // [4,8,128,128]
  const float* w_mix   = (const float*)d_in[2];  // [1024,128]
  const float* b_mix   = (const float*)d_in[3];  // [128]
  float* out = (float*)d_out;                    // [8,8192,128]

  char* ws = (char*)d_ws;
  float* ping     = (float*)ws;                                   // 32 MiB
  float* weightsT = (float*)(ws + (size_t)RTOT * DD * 4);         // 2 MiB
  float* w_mixT   = (float*)(ws + (size_t)RTOT * DD * 4 + 2097152); // 0.5 MiB

  fractal_prep_transpose<<<2048, 256, 0, stream>>>(weights, w_mix, weightsT, w_mixT);

  const int jobs = RTOT / 16;  // 4096 workgroups per level
  // level k=1..4: L = 1024<<lev, weights[lev]; ping-pong through ws/out
  fractal_level_kernel<<<jobs, 256, 0, stream>>>(x,    weightsT + 0 * 8 * 16384,
                                                 w_mixT, b_mix, ping, 1023, 10);
  fractal_level_kernel<<<jobs, 256, 0, stream>>>(ping, weightsT + 1 * 8 * 16384,
                                                 w_mixT, b_mix, out,  2047, 11);
  fractal_level_kernel<<<jobs, 256, 0, stream>>>(out,  weightsT + 2 * 8 * 16384,
                                                 w_mixT, b_mix, ping, 4095, 12);
  fractal_level_kernel<<<jobs, 256, 0, stream>>>(ping, weightsT + 3 * 8 * 16384,
                                                 w_mixT, b_mix, out,  8191, 13);
}